// HetGCN_3_17884243821060
// MI455X (gfx1250) — compile-verified
//
#include <hip/hip_runtime.h>
#include <hip/hip_bf16.h>
#include <math.h>

typedef __attribute__((ext_vector_type(2))) float v2f;
typedef __attribute__((ext_vector_type(8))) float v8f;

#define T_TYPES 7
#define H_DIM   16
#define TH      112   // T_TYPES * H_DIM
#define MID     32

// ---------------------------------------------------------------------------
// Per-node, per-type linear: z[n, :16] = x[n, :F] @ Wt[type[n], :, :]
// One thread per (node, h) output element. Tiny FLOPs; memory/latency bound.
// ---------------------------------------------------------------------------
template<int F>
__global__ void het_transform(const float* __restrict__ x,     // [N, F]
                              const int*   __restrict__ ntype, // [N]
                              const float* __restrict__ Wt,    // [T, F, 16]
                              float* __restrict__ z, int n) {
  int i = blockIdx.x * blockDim.x + threadIdx.x;
  if (i >= n * H_DIM) return;
  int node = i >> 4, h = i & 15;
  int t = ntype[node];
  const float* w  = Wt + ((size_t)t * F) * H_DIM + h;
  const float* xv = x + (size_t)node * F;
  float acc = 0.f;
#pragma unroll
  for (int f = 0; f < F; ++f) acc = fmaf(xv[f], w[f * H_DIM], acc);
  z[(size_t)node * H_DIM + h] = acc;
}

// ---------------------------------------------------------------------------
// Edge scatter: agg[(dst*T + type[src]) * 16 + h] += w[e] * z[src*16 + h]
// 16 lanes per edge; fp32 global atomics (land in L2 — agg fits).
// ---------------------------------------------------------------------------
__global__ void het_scatter(const int*   __restrict__ src,
                            const int*   __restrict__ dst,
                            const float* __restrict__ ew,
                            const int*   __restrict__ ntype,
                            const float* __restrict__ z,     // [N, 16]
                            float* __restrict__ agg,         // [N, T, 16]
                            int e) {
  long long i = (long long)blockIdx.x * blockDim.x + threadIdx.x;
  if (i >= (long long)e * H_DIM) return;
  int eidx = (int)(i >> 4);
  int h    = (int)(i & 15);
  int s = src[eidx], d = dst[eidx];
  int t = ntype[s];
  float m = ew[eidx] * z[(size_t)s * H_DIM + h];
  unsafeAtomicAdd(&agg[((size_t)d * T_TYPES + t) * H_DIM + h], m);
}

// ---------------------------------------------------------------------------
// Out-linear: [N,112] @ [112,32] + b via V_WMMA_F32_16X16X4_F32.
// One wave owns a 16-node x 32-col tile: 2 v8f accumulators, 28 K-steps of 4.
// MODE 0: leaky_relu -> store h [N,32].
// MODE 1: sigmoid -> LDS block reduction -> atomic add into out[32].
// ---------------------------------------------------------------------------
template<int MODE>
__global__ void het_out_linear(const float* __restrict__ agg, // [N, 112]
                               const float* __restrict__ Wo,  // [112, 32]
                               const float* __restrict__ b,   // [32]
                               float* __restrict__ out,       // [N,32] or [32]
                               int n) {
  __shared__ float sWo[TH * MID];
  __shared__ float sB[MID];
  __shared__ float sSum[MID];

  for (int i = threadIdx.x; i < TH * MID; i += blockDim.x) sWo[i] = Wo[i];
  if (threadIdx.x < MID) {
    sB[threadIdx.x] = b[threadIdx.x];
    if (MODE) sSum[threadIdx.x] = 0.f;
  }
  __syncthreads();

  int wave = threadIdx.x >> 5;
  int lane = threadIdx.x & 31;
  int ntiles = n >> 4;                       // N is a multiple of 16
  int tile = blockIdx.x * (blockDim.x >> 5) + wave;

  if (tile < ntiles) {                       // wave-uniform: EXEC all-1s inside
    int node0 = tile << 4;
    int m    = lane & 15;                    // M (for A) / N (for B,D)
    int koff = (lane >> 4) << 1;             // lanes 0-15: K=0,1; lanes 16-31: K=2,3
    const float* arow = agg + (size_t)(node0 + m) * TH;

    v8f acc0 = {}; v8f acc1 = {};
#pragma unroll
    for (int k = 0; k < TH; k += 4) {
      v2f a;  a.x  = arow[k + koff];              a.y  = arow[k + koff + 1];
      v2f b0; b0.x = sWo[(k + koff) * MID + m];   b0.y = sWo[(k + koff + 1) * MID + m];
      v2f b1; b1.x = sWo[(k + koff) * MID + 16 + m];
              b1.y = sWo[(k + koff + 1) * MID + 16 + m];
      acc0 = __builtin_amdgcn_wmma_f32_16x16x4_f32(false, a, false, b0,
                                                   (short)0, acc0, false, false);
      acc1 = __builtin_amdgcn_wmma_f32_16x16x4_f32(false, a, false, b1,
                                                   (short)0, acc1, false, false);
    }

    int mbase = (lane >> 4) << 3;            // D: lanes>=16 hold rows M=r+8
#pragma unroll
    for (int r = 0; r < 8; ++r) {
      int row  = node0 + mbase + r;
      float v0 = acc0[r] + sB[m];
      float v1 = acc1[r] + sB[16 + m];
      if (MODE == 0) {
        v0 = v0 > 0.f ? v0 : 0.01f * v0;     // leaky_relu(0.01)
        v1 = v1 > 0.f ? v1 : 0.01f * v1;
        out[(size_t)row * MID + m]      = v0;
        out[(size_t)row * MID + 16 + m] = v1;
      } else {
        v0 = 1.f / (1.f + __expf(-v0));      // sigmoid
        v1 = 1.f / (1.f + __expf(-v1));
        atomicAdd(&sSum[m],      v0);        // ds_add_f32
        atomicAdd(&sSum[16 + m], v1);
      }
    }
  }

  if (MODE) {
    __syncthreads();
    if (threadIdx.x < MID) unsafeAtomicAdd(&out[threadIdx.x], sSum[threadIdx.x]);
  }
}

// ---------------------------------------------------------------------------
extern "C" void kernel_launch(void* const* d_in, const int* in_sizes, int n_in,
                              void* d_out, int out_size, void* d_ws, size_t ws_size,
                              hipStream_t stream) {
  const float* x    = (const float*)d_in[0];  // [N, 7]
  const float* ew   = (const float*)d_in[1];  // [E]
  const float* W1   = (const float*)d_in[2];  // [7, 7, 16]
  const float* Wo1  = (const float*)d_in[3];  // [112, 32]
  const float* b1   = (const float*)d_in[4];  // [32]
  const float* W2   = (const float*)d_in[5];  // [7, 32, 16]
  const float* Wo2  = (const float*)d_in[6];  // [112, 32]
  const float* b2   = (const float*)d_in[7];  // [32]
  const int*   eidx = (const int*)d_in[8];    // [2, E]
  const int*   ntyp = (const int*)d_in[9];    // [N]

  int n = in_sizes[0] / 7;                    // F_IN = 7
  int e = in_sizes[1];
  const int* src = eidx;
  const int* dst = eidx + e;

  // Workspace: agg [N*112] | z [N*16] | h1 [N*32]  (~64 MB total)
  float* agg = (float*)d_ws;
  float* z   = agg + (size_t)n * TH;
  float* h1  = z + (size_t)n * H_DIM;

  const int thr = 256;
  int gT = (n * H_DIM + thr - 1) / thr;
  long long scat = (long long)e * H_DIM;
  int gS = (int)((scat + thr - 1) / thr);
  int wpb = thr / 32;
  int gL = ((n >> 4) + wpb - 1) / wpb;

  // ---- Layer 1 ----
  hipMemsetAsync(agg, 0, (size_t)n * TH * sizeof(float), stream);
  het_transform<7><<<gT, thr, 0, stream>>>(x, ntyp, W1, z, n);
  het_scatter<<<gS, thr, 0, stream>>>(src, dst, ew, ntyp, z, agg, e);
  het_out_linear<0><<<gL, thr, 0, stream>>>(agg, Wo1, b1, h1, n);

  // ---- Layer 2 ----
  hipMemsetAsync(agg, 0, (size_t)n * TH * sizeof(float), stream);
  het_transform<32><<<gT, thr, 0, stream>>>(h1, ntyp, W2, z, n);
  het_scatter<<<gS, thr, 0, stream>>>(src, dst, ew, ntyp, z, agg, e);
  hipMemsetAsync(d_out, 0, (size_t)out_size * sizeof(float), stream);
  het_out_linear<1><<<gL, thr, 0, stream>>>(agg, Wo2, b2, (float*)d_out, n);
}